// FactorTripletGCN_19000935317643
// MI455X (gfx1250) — compile-verified
//
#include <hip/hip_runtime.h>
#include <hip/hip_bf16.h>
#include <math.h>

typedef __attribute__((ext_vector_type(2))) float v2f;
typedef __attribute__((ext_vector_type(8))) float v8f;

static inline int cdiv_i(long long a, int b) { return (int)((a + b - 1) / b); }

// ---------------------------------------------------------------------------
// fp32 GEMM on the matrix pipe: C[M x N] (+)= A[M x K] @ B[K x N] (+ bias)
// V_WMMA_F32_16X16X4_F32. Wave tile 32(M) x 64(N) (2 A-frags x 4 B-frags,
// 8 accumulators) so the 8 B loads per K-step feed 8 WMMAs. 8 waves stacked
// in M -> block tile 256 x 64. All shapes here: M % 256 == 0, N % 64 == 0,
// K % 16 == 0.
// ---------------------------------------------------------------------------
template<bool ACC, bool BIAS>
__global__ __launch_bounds__(256) void gemm_f32_wmma(
    const float* __restrict__ A, const float* __restrict__ B,
    const float* __restrict__ bias, float* __restrict__ C,
    int M, int K, int N)
{
  const int wave  = threadIdx.x >> 5;   // 0..7
  const int lane  = threadIdx.x & 31;
  const int r16   = lane & 15;
  const int khalf = lane >> 4;          // 0 or 1
  const int m0 = blockIdx.x * 256 + wave * 32;
  const int n0 = blockIdx.y * 64;
  if (m0 >= M) return;                  // wave-uniform

  v8f acc[2][4] = {};
  const float* __restrict__ arow0 = A + (size_t)(m0 + r16) * K;
  const float* __restrict__ arow1 = A + (size_t)(m0 + 16 + r16) * K;
  #pragma unroll 4
  for (int k0 = 0; k0 < K; k0 += 4) {
    const int ka = k0 + 2 * khalf;
    // A 16x4 fragment: lane = row, VGPR0 = K {0|2}, VGPR1 = K {1|3}
    v2f a0; a0.x = arow0[ka]; a0.y = arow0[ka + 1];
    v2f a1; a1.x = arow1[ka]; a1.y = arow1[ka + 1];
    const float* __restrict__ bp = B + (size_t)ka * N + n0 + r16;
    v2f b[4];
    #pragma unroll
    for (int nf = 0; nf < 4; ++nf) { b[nf].x = bp[nf * 16]; b[nf].y = bp[N + nf * 16]; }
    #pragma unroll
    for (int nf = 0; nf < 4; ++nf) {
      acc[0][nf] = __builtin_amdgcn_wmma_f32_16x16x4_f32(false, a0, false, b[nf], (short)0, acc[0][nf], false, false);
      acc[1][nf] = __builtin_amdgcn_wmma_f32_16x16x4_f32(false, a1, false, b[nf], (short)0, acc[1][nf], false, false);
    }
  }
  // C/D layout: VGPR r holds row (r + 8*khalf), column = n-tile + (lane&15)
  #pragma unroll
  for (int mf = 0; mf < 2; ++mf) {
    #pragma unroll
    for (int nf = 0; nf < 4; ++nf) {
      #pragma unroll
      for (int r = 0; r < 8; ++r) {
        const int row = m0 + mf * 16 + r + 8 * khalf;
        const int col = n0 + nf * 16 + r16;
        const size_t off = (size_t)row * N + col;
        float v = acc[mf][nf][r];
        if (BIAS) v += bias[col];
        if (ACC)  v += C[off];
        C[off] = v;
      }
    }
  }
}

// ---------------------------------------------------------------------------
// Elementwise / scatter kernels
// ---------------------------------------------------------------------------
__global__ void k_build_tvecs(const float* __restrict__ obj, const float* __restrict__ pred,
                              const int* __restrict__ edges, float* __restrict__ tv, int T)
{
  const int n = T * 384;
  for (int idx = blockIdx.x * blockDim.x + threadIdx.x; idx < n; idx += gridDim.x * blockDim.x) {
    const int e = idx / 384, c = idx - e * 384;
    const int s = edges[2 * e], d = edges[2 * e + 1];
    float v;
    if (c < 128)      v = obj[(size_t)s * 128 + c];
    else if (c < 256) v = pred[(size_t)e * 128 + (c - 128)];
    else              v = obj[(size_t)d * 128 + (c - 256)];
    tv[idx] = v;
  }
}

__global__ void k_deg(const int* __restrict__ edges, float* __restrict__ deg, int T)
{
  for (int e = blockIdx.x * blockDim.x + threadIdx.x; e < T; e += gridDim.x * blockDim.x)
    atomicAdd(&deg[edges[2 * e + 1]], 1.0f);
}

__global__ void k_norm(float* __restrict__ deg, int N)
{
  for (int i = blockIdx.x * blockDim.x + threadIdx.x; i < N; i += gridDim.x * blockDim.x)
    deg[i] = rsqrtf(fmaxf(deg[i], 1.0f));
}

__global__ void k_logit_init(float* __restrict__ logit,
                             const float* b0, const float* b1, const float* b2,
                             const float* b3, const float* b4, const float* b5,
                             int N, int C)
{
  const long long n = (long long)N * C;
  for (long long idx = blockIdx.x * (long long)blockDim.x + threadIdx.x; idx < n;
       idx += (long long)gridDim.x * blockDim.x) {
    const int c = (int)(idx % C);
    logit[idx] = b0[c] + b1[c] + b2[c] + b3[c] + b4[c] + b5[c];
  }
}

__global__ void k_att(const float* __restrict__ h,
                      const float* __restrict__ Wl, const float* __restrict__ bl,
                      const float* __restrict__ Wr, const float* __restrict__ br,
                      float* __restrict__ aL, float* __restrict__ aR,
                      int N, int F, int nl)
{
  const int n = N * nl;
  for (int idx = blockIdx.x * blockDim.x + threadIdx.x; idx < n; idx += gridDim.x * blockDim.x) {
    const int r = idx / nl, k = idx - r * nl;
    const float* __restrict__ hr = h + (size_t)r * F;
    float sL = bl[k], sR = br[k];
    for (int f = 0; f < F; ++f) {
      const float hv = hr[f];
      sL = fmaf(hv, Wl[f * nl + k], sL);
      sR = fmaf(hv, Wr[f * nl + k], sR);
    }
    aL[idx] = sL; aR[idx] = sR;
  }
}

__global__ void k_scale_rows(float* __restrict__ h, const float* __restrict__ nrm, int N, int F)
{
  const long long n = (long long)N * F;
  for (long long idx = blockIdx.x * (long long)blockDim.x + threadIdx.x; idx < n;
       idx += (long long)gridDim.x * blockDim.x)
    h[idx] *= nrm[idx / F];
}

// one 512-thread block per edge; 512 = nl*F always
__global__ __launch_bounds__(512) void k_scatter_msg(
    const float* __restrict__ h, const float* __restrict__ aL, const float* __restrict__ aR,
    const int* __restrict__ edges, float* __restrict__ out, int F, int nl)
{
  const int e = blockIdx.x;
  const int j = threadIdx.x;
  const int s = edges[2 * e], d = edges[2 * e + 1];
  const int k = j / F, f = j - k * F;
  const float x = 6.0f * (aL[(size_t)s * nl + k] + aR[(size_t)d * nl + k]);
  const float fac = 1.0f / (1.0f + __expf(-x));
  atomicAdd(&out[(size_t)d * 512 + j], h[(size_t)s * F + f] * fac);
}

__global__ __launch_bounds__(512) void k_bn_stats(const float* __restrict__ x,
                                                  float* __restrict__ stats, int N)
{
  const int c = threadIdx.x;         // channel 0..511
  float s = 0.f, sq = 0.f;
  for (int r = blockIdx.x; r < N; r += gridDim.x) {
    const float v = x[(size_t)r * 512 + c];
    s += v; sq += v * v;
  }
  atomicAdd(&stats[c], s);
  atomicAdd(&stats[512 + c], sq);
}

__global__ __launch_bounds__(512) void k_bn_finalize(float* __restrict__ stats,
                                                     const float* __restrict__ gamma,
                                                     const float* __restrict__ beta, int N)
{
  const int c = threadIdx.x;
  const float invN = 1.0f / (float)N;
  const float mu  = stats[c] * invN;
  const float var = stats[512 + c] * invN - mu * mu;
  const float sc  = gamma[c] * rsqrtf(var + 1e-5f);
  stats[1024 + c] = sc;
  stats[1536 + c] = beta[c] - mu * sc;
}

__global__ void k_bn_apply_relu(float* __restrict__ x, const float* __restrict__ stats, int N)
{
  const long long n = (long long)N * 512;
  for (long long idx = blockIdx.x * (long long)blockDim.x + threadIdx.x; idx < n;
       idx += (long long)gridDim.x * blockDim.x) {
    const int c = (int)(idx & 511);
    const float v = x[idx] * stats[1024 + c] + stats[1536 + c];
    x[idx] = fmaxf(v, 0.0f);
  }
}

__global__ void k_copy_pred(const float* __restrict__ L, float* __restrict__ out, int T)
{
  const int n = T * 128;
  for (int idx = blockIdx.x * blockDim.x + threadIdx.x; idx < n; idx += gridDim.x * blockDim.x) {
    const int e = idx >> 7, c = idx & 127;
    out[idx] = L[(size_t)e * 384 + 128 + c];
  }
}

__global__ void k_pool_scatter(const float* __restrict__ L, const int* __restrict__ edges,
                               float* __restrict__ pooled, int T)
{
  const int n = T * 128;
  for (int idx = blockIdx.x * blockDim.x + threadIdx.x; idx < n; idx += gridDim.x * blockDim.x) {
    const int e = idx >> 7, c = idx & 127;
    const int s = edges[2 * e], d = edges[2 * e + 1];
    atomicAdd(&pooled[(size_t)s * 128 + c], L[(size_t)e * 384 + c]);
    atomicAdd(&pooled[(size_t)d * 128 + c], L[(size_t)e * 384 + 256 + c]);
  }
}

__global__ void k_cnt_scatter(const int* __restrict__ edges, float* __restrict__ cnt, int T)
{
  for (int e = blockIdx.x * blockDim.x + threadIdx.x; e < T; e += gridDim.x * blockDim.x) {
    atomicAdd(&cnt[edges[2 * e]], 1.0f);
    atomicAdd(&cnt[edges[2 * e + 1]], 1.0f);
  }
}

__global__ void k_pool_norm(float* __restrict__ pooled, const float* __restrict__ cnt, int O)
{
  const int n = O * 128;
  for (int idx = blockIdx.x * blockDim.x + threadIdx.x; idx < n; idx += gridDim.x * blockDim.x)
    pooled[idx] /= fmaxf(cnt[idx >> 7], 1.0f);
}

// ---------------------------------------------------------------------------
// Host-side parameter mapping (JAX tree_leaves order: sorted dict keys)
//   'bns' (beta,gamma x5) -> 'layers' (att_l.{W,b}, att_r.{W,b}, lin.{W,b} x5)
//   -> 'linears' ({W,b} x6)
// ---------------------------------------------------------------------------
struct Lin { const float* W; const float* b; };
struct Net {
  const float* bn_beta[5]; const float* bn_gamma[5];
  Lin att_l[5], att_r[5], lin[5];
  Lin linears[6];
};

static Net parse_net(void* const* d_in, int base) {
  Net n; int i = base;
  for (int l = 0; l < 5; ++l) { n.bn_beta[l] = (const float*)d_in[i++]; n.bn_gamma[l] = (const float*)d_in[i++]; }
  for (int l = 0; l < 5; ++l) {
    n.att_l[l].W = (const float*)d_in[i++]; n.att_l[l].b = (const float*)d_in[i++];
    n.att_r[l].W = (const float*)d_in[i++]; n.att_r[l].b = (const float*)d_in[i++];
    n.lin[l].W   = (const float*)d_in[i++]; n.lin[l].b   = (const float*)d_in[i++];
  }
  for (int l = 0; l < 6; ++l) { n.linears[l].W = (const float*)d_in[i++]; n.linears[l].b = (const float*)d_in[i++]; }
  return n;
}

static void run_gnn(hipStream_t stream, const Net& net,
                    const float* x, int in_dim,
                    const int* edges, int T, int N, int C, float* logit,
                    float* bufA, float* bufH, float* aL, float* aR,
                    float* nrm, float* stats)
{
  static const int F_arr[5]  = {64, 64, 128, 128, 256};
  static const int nl_arr[5] = {8, 8, 4, 4, 2};
  const dim3 blk(256);

  hipMemsetAsync(nrm, 0, (size_t)N * 4, stream);
  k_deg <<<cdiv_i(T, 256), 256, 0, stream>>>(edges, nrm, T);
  k_norm<<<cdiv_i(N, 256), 256, 0, stream>>>(nrm, N);

  k_logit_init<<<cdiv_i((long long)N * C, 256), 256, 0, stream>>>(
      logit, net.linears[0].b, net.linears[1].b, net.linears[2].b,
      net.linears[3].b, net.linears[4].b, net.linears[5].b, N, C);

  { dim3 g(cdiv_i(N, 256), C / 64);
    gemm_f32_wmma<true, false><<<g, blk, 0, stream>>>(x, net.linears[0].W, nullptr, logit, N, in_dim, C); }

  const float* feat = x; int fdim = in_dim;
  for (int l = 0; l < 5; ++l) {
    const int F = F_arr[l], nl = nl_arr[l];
    { dim3 g(cdiv_i(N, 256), F / 64);
      gemm_f32_wmma<false, true><<<g, blk, 0, stream>>>(feat, net.lin[l].W, net.lin[l].b, bufH, N, fdim, F); }
    k_att<<<cdiv_i(N * nl, 256), 256, 0, stream>>>(bufH, net.att_l[l].W, net.att_l[l].b,
                                                   net.att_r[l].W, net.att_r[l].b, aL, aR, N, F, nl);
    k_scale_rows<<<cdiv_i((long long)N * F, 256), 256, 0, stream>>>(bufH, nrm, N, F);
    hipMemsetAsync(bufA, 0, (size_t)N * 512 * 4, stream);
    k_scatter_msg<<<T, 512, 0, stream>>>(bufH, aL, aR, edges, bufA, F, nl);
    hipMemsetAsync(stats, 0, 1024 * 4, stream);
    k_bn_stats   <<<256, 512, 0, stream>>>(bufA, stats, N);
    k_bn_finalize<<<1, 512, 0, stream>>>(stats, net.bn_gamma[l], net.bn_beta[l], N);
    k_bn_apply_relu<<<cdiv_i((long long)N * 512, 256), 256, 0, stream>>>(bufA, stats, N);
    { dim3 g(cdiv_i(N, 256), C / 64);
      gemm_f32_wmma<true, false><<<g, blk, 0, stream>>>(bufA, net.linears[l + 1].W, nullptr, logit, N, 512, C); }
    feat = bufA; fdim = 512;
  }
}

extern "C" void kernel_launch(void* const* d_in, const int* in_sizes, int n_in,
                              void* d_out, int out_size, void* d_ws, size_t ws_size,
                              hipStream_t stream)
{
  (void)in_sizes; (void)n_in; (void)out_size; (void)ws_size;
  const int O = 32000, T = 64000;

  const float* obj  = (const float*)d_in[0];
  const float* pred = (const float*)d_in[1];
  const Net net1 = parse_net(d_in, 2);
  const Net net2 = parse_net(d_in, 54);
  const int* edges = (const int*)d_in[106];

  float* w = (float*)d_ws;
  float* bufA   = w; w += (long long)64000 * 512;   // feat / t_vecs / msg accumulator
  float* bufH   = w; w += (long long)64000 * 256;   // hidden
  float* bufL   = w; w += (long long)64000 * 384;   // net1 logits
  float* aL     = w; w += (long long)64000 * 8;
  float* aR     = w; w += (long long)64000 * 8;
  float* nrm    = w; w += 64000;
  float* pooled = w; w += (long long)32000 * 128;
  float* cnt    = w; w += 32000;
  float* stats  = w; w += 2048;

  float* outF    = (float*)d_out;
  float* obj_out = outF;                      // 32000*128
  float* pred_out = outF + (size_t)O * 128;   // 64000*128

  // --- net1 on triplets -------------------------------------------------
  k_build_tvecs<<<cdiv_i((long long)T * 384, 256), 256, 0, stream>>>(obj, pred, edges, bufA, T);
  run_gnn(stream, net1, bufA, 384, edges, T, /*N=*/T, /*C=*/384, bufL,
          bufA, bufH, aL, aR, nrm, stats);

  k_copy_pred<<<cdiv_i((long long)T * 128, 256), 256, 0, stream>>>(bufL, pred_out, T);

  // --- pool new_s / new_o back onto objects -----------------------------
  hipMemsetAsync(pooled, 0, (size_t)O * 128 * 4, stream);
  hipMemsetAsync(cnt, 0, (size_t)O * 4, stream);
  k_pool_scatter<<<cdiv_i((long long)T * 128, 256), 256, 0, stream>>>(bufL, edges, pooled, T);
  k_cnt_scatter <<<cdiv_i(T, 256), 256, 0, stream>>>(edges, cnt, T);
  k_pool_norm   <<<cdiv_i((long long)O * 128, 256), 256, 0, stream>>>(pooled, cnt, O);

  // --- net2 on objects, logits written straight into d_out --------------
  run_gnn(stream, net2, pooled, 128, edges, T, /*N=*/O, /*C=*/128, obj_out,
          bufA, bufH, aL, aR, nrm, stats);
}